// TinyQuantumClassifier_49486613184839
// MI455X (gfx1250) — compile-verified
//
#include <hip/hip_runtime.h>

// ---------------------------------------------------------------------------
// TinyQuantumClassifier for MI455X (gfx1250, wave32)
//
// Math reduction: z(x1,x2) = u(x1)^T A v(x2),  u = (1, cos x1, sin x1),
// v = (1, cos x2, sin x2), where the 3x3 A is derived once per launch from
// theta:  V = C*G2*C*G1 (4x4),  M = V^T D V,  then half-angle identities fold
// the quadratic form M into A.  The two 4x4 matmuls are done with
// v_wmma_f32_16x16x4_f32 (zero-padded 16x4 / 4x16 operands) in a 1-wave
// setup kernel; the 8.4M-sample batch kernel is a pure streaming kernel
// (2x global_load_b128 + 8 sincos pairs + 32 FMA + 1x global_store_b128
// per thread), which sits on the 23.3 TB/s HBM roofline (~100 MB traffic).
// ---------------------------------------------------------------------------

typedef float v2f __attribute__((ext_vector_type(2)));
typedef float v8f __attribute__((ext_vector_type(8)));

#define INV_2PI 0.15915494309189535f  // v_sin_f32/v_cos_f32 take "turns"

__device__ __forceinline__ float lane_gather16(float v, int srcLane) {
  // wave32 ds_bpermute: dst[lane] = src[srcLane]; byte index
  int r = __builtin_amdgcn_ds_bpermute(srcLane << 2, __builtin_bit_cast(int, v));
  return __builtin_bit_cast(float, r);
}

// ---------------------------------------------------------------------------
// Setup: one wave. Builds the 3x3 coefficient matrix A (9 floats) in d_ws.
// Uses two V_WMMA_F32_16X16X4_F32 ops for the 4x4 circuit-matrix products.
// ---------------------------------------------------------------------------
__global__ __launch_bounds__(32) void tqc_setup(const float* __restrict__ theta,
                                                float* __restrict__ coeff) {
  __shared__ float K2s[16];  // C * G2 * C
  __shared__ float G1s[16];  // A1 (x) B1
  __shared__ float Mm[16];   // V^T D V

  const int lane = threadIdx.x;

  if (lane == 0) {
    float ang[4] = {theta[0], theta[1], theta[2], theta[3]};
    float c[4], s[4];
#pragma unroll
    for (int k = 0; k < 4; ++k) {
      float f = ang[k] * (0.5f * INV_2PI);  // half angle, in turns
      s[k] = __builtin_amdgcn_sinf(f);
      c[k] = __builtin_amdgcn_cosf(f);
    }
    // RY 2x2 gates: layer1 = (wire0: idx0, wire1: idx1), layer2 = (idx2, idx3)
    float A1[2][2] = {{c[0], -s[0]}, {s[0], c[0]}};
    float B1[2][2] = {{c[1], -s[1]}, {s[1], c[1]}};
    float A2[2][2] = {{c[2], -s[2]}, {s[2], c[2]}};
    float B2[2][2] = {{c[3], -s[3]}, {s[3], c[3]}};
    // G[(2p+q)][(2r+t)] = A[p][r]*B[q][t]  (row-major vec of 2x2 state S[p][q])
    float G2[16];
#pragma unroll
    for (int p = 0; p < 2; ++p)
#pragma unroll
      for (int q = 0; q < 2; ++q)
#pragma unroll
        for (int r = 0; r < 2; ++r)
#pragma unroll
          for (int t = 0; t < 2; ++t) {
            G1s[(2 * p + q) * 4 + (2 * r + t)] = A1[p][r] * B1[q][t];
            G2[(2 * p + q) * 4 + (2 * r + t)] = A2[p][r] * B2[q][t];
          }
    // K2 = C * G2 * C  (CNOT(0,1) swaps vec components 2 and 3)
    const int sw[4] = {0, 1, 3, 2};
#pragma unroll
    for (int i = 0; i < 4; ++i)
#pragma unroll
      for (int j = 0; j < 4; ++j)
        K2s[i * 4 + j] = G2[sw[i] * 4 + sw[j]];
  }
  __syncthreads();

  const int row = lane & 15;
  const int hi = lane >> 4;

  // --- WMMA #1: V = K2 * G1 (4x4 result in top-left of 16x16) ---
  // A layout (16x4 f32): lanes 0-15 hold K=0,1; lanes 16-31 hold K=2,3.
  // B layout (4x16 f32): vgpr0 = rows 0/2 (low/high lanes), vgpr1 = rows 1/3.
  v2f a1, b1;
  a1.x = (row < 4) ? K2s[row * 4 + (hi ? 2 : 0)] : 0.0f;
  a1.y = (row < 4) ? K2s[row * 4 + (hi ? 3 : 1)] : 0.0f;
  b1.x = (row < 4) ? G1s[(hi ? 2 : 0) * 4 + row] : 0.0f;
  b1.y = (row < 4) ? G1s[(hi ? 3 : 1) * 4 + row] : 0.0f;
  v8f zero8 = {};
  v8f d1 = __builtin_amdgcn_wmma_f32_16x16x4_f32(
      /*neg_a=*/false, a1, /*neg_b=*/false, b1,
      /*c_mod=*/(short)0, zero8, /*reuse_a=*/false, /*reuse_b=*/false);

  // D layout: d1[k] at lane n (n<16) = V[k][n].  Gather V[k][row] to all lanes.
  float t0 = lane_gather16(d1[0], row);
  float t1 = lane_gather16(d1[1], row);
  float t2 = lane_gather16(d1[2], row);
  float t3 = lane_gather16(d1[3], row);

  // --- WMMA #2: M = (V^T) * (D*V),  D = diag(1,1,-1,-1) ---
  v2f a2, b2;
  a2.x = hi ? t2 : t0;   // A2[row][k] = V[k][row], k = hi?2:0
  a2.y = hi ? t3 : t1;   //                         k = hi?3:1
  b2.x = hi ? -t2 : t0;  // B2[k][row] = sign_k * V[k][row]
  b2.y = hi ? -t3 : t1;
  v8f d2 = __builtin_amdgcn_wmma_f32_16x16x4_f32(
      false, a2, false, b2, (short)0, zero8, false, false);

  // d2[p] at lane q (p,q < 4) = M[p][q]
  if (lane < 4) {
    Mm[0 * 4 + lane] = d2[0];
    Mm[1 * 4 + lane] = d2[1];
    Mm[2 * 4 + lane] = d2[2];
    Mm[3 * 4 + lane] = d2[3];
  }
  __syncthreads();

  if (lane == 0) {
    // Half-angle fold: v_p v_r -> basis (1, cos, sin):
    //   v0^2 = (1+cos)/2, v1^2 = (1-cos)/2, v0 v1 = sin/2
    const float T[2][2][3] = {{{0.5f, 0.5f, 0.0f}, {0.0f, 0.0f, 0.5f}},
                              {{0.0f, 0.0f, 0.5f}, {0.5f, -0.5f, 0.0f}}};
#pragma unroll
    for (int i = 0; i < 3; ++i)
#pragma unroll
      for (int j = 0; j < 3; ++j) {
        float acc = 0.0f;
#pragma unroll
        for (int p = 0; p < 2; ++p)
#pragma unroll
          for (int r = 0; r < 2; ++r)
#pragma unroll
            for (int q = 0; q < 2; ++q)
#pragma unroll
              for (int t = 0; t < 2; ++t)
                acc += T[p][r][i] * T[q][t][j] * Mm[(2 * p + q) * 4 + (2 * r + t)];
        coeff[i * 3 + j] = acc;
      }
  }
}

// ---------------------------------------------------------------------------
// Batch kernel: 4 samples per thread.  Streaming, HBM-roofline bound.
// ---------------------------------------------------------------------------
__global__ __launch_bounds__(256) void tqc_main(const float4* __restrict__ feat,
                                                const float* __restrict__ coeff,
                                                float4* __restrict__ out,
                                                int nquad) {
  const int t = blockIdx.x * blockDim.x + threadIdx.x;
  if (t >= nquad) return;

  const float a00 = coeff[0], a01 = coeff[1], a02 = coeff[2];
  const float a10 = coeff[3], a11 = coeff[4], a12 = coeff[5];
  const float a20 = coeff[6], a21 = coeff[7], a22 = coeff[8];

  const float4 fa = feat[2 * t];      // samples 4t, 4t+1
  const float4 fb = feat[2 * t + 1];  // samples 4t+2, 4t+3

  auto ev = [&](float x1, float x2) -> float {
    float f1 = x1 * INV_2PI;
    float f2 = x2 * INV_2PI;
    float c1 = __builtin_amdgcn_cosf(f1);
    float s1 = __builtin_amdgcn_sinf(f1);
    float c2 = __builtin_amdgcn_cosf(f2);
    float s2 = __builtin_amdgcn_sinf(f2);
    float w0 = __builtin_fmaf(a02, s2, __builtin_fmaf(a01, c2, a00));
    float w1 = __builtin_fmaf(a12, s2, __builtin_fmaf(a11, c2, a10));
    float w2 = __builtin_fmaf(a22, s2, __builtin_fmaf(a21, c2, a20));
    return __builtin_fmaf(s1, w2, __builtin_fmaf(c1, w1, w0));
  };

  float4 z;
  z.x = ev(fa.x, fa.y);
  z.y = ev(fa.z, fa.w);
  z.z = ev(fb.x, fb.y);
  z.w = ev(fb.z, fb.w);
  out[t] = z;
}

extern "C" void kernel_launch(void* const* d_in, const int* in_sizes, int n_in,
                              void* d_out, int out_size, void* d_ws, size_t ws_size,
                              hipStream_t stream) {
  const float* feat = (const float*)d_in[0];   // [B,2] float32
  const float* theta = (const float*)d_in[1];  // [2,2] float32
  float* out = (float*)d_out;                  // [B,1] float32
  float* coeff = (float*)d_ws;                 // 9 floats of scratch

  // 1) Fold theta -> 3x3 bilinear coefficient matrix (one wave, 2x WMMA).
  tqc_setup<<<1, 32, 0, stream>>>(theta, coeff);

  // 2) Stream the batch: 4 samples per thread (B = 8388608 is divisible by 4).
  const int B = in_sizes[0] / 2;
  const int nquad = B / 4;
  const int blocks = (nquad + 255) / 256;
  tqc_main<<<blocks, 256, 0, stream>>>((const float4*)feat, coeff,
                                       (float4*)out, nquad);
}